// GraphConvLayer_70669391888427
// MI455X (gfx1250) — compile-verified
//
#include <hip/hip_runtime.h>

typedef float v2f __attribute__((ext_vector_type(2)));
typedef float v8f __attribute__((ext_vector_type(8)));

#define N_NODES 65536
#define N_EDGES 1048576
#define D 64   // D_IN == D_OUT == 64

// ---------------------------------------------------------------------------
// Kernel 1: out[n, d] = bias[d]   (segment_sum accumulates on top of this)
// ---------------------------------------------------------------------------
__global__ void gcn_init_bias(const float* __restrict__ bias,
                              float* __restrict__ out) {
    int i = blockIdx.x * blockDim.x + threadIdx.x;   // over N_NODES * D
    out[i] = bias[i & (D - 1)];
}

// ---------------------------------------------------------------------------
// Kernel 2: h = support @ weight  via V_WMMA_F32_16X16X4_F32 (exact fp32).
// One wave per 16x16 output tile; K=64 -> 16 WMMA steps of K=4.
//
// f32 A 16x4 layout : lane L(0..15) v0=A[L][k+0], v1=A[L][k+1];
//                     lane L+16     v0=A[L][k+2], v1=A[L][k+3]
// f32 B 4x16 layout : v0 lanes0-15 = row k+0, lanes16-31 = row k+2;
//                     v1 lanes0-15 = row k+1, lanes16-31 = row k+3
// f32 C/D 16x16     : VGPR r: lanes0-15 M=r (N=lane), lanes16-31 M=r+8
// ---------------------------------------------------------------------------
__global__ void gcn_gemm_wmma(const float* __restrict__ A,   // [N_NODES, D]
                              const float* __restrict__ B,   // [D, D]
                              float* __restrict__ H) {       // [N_NODES, D]
    const int lane  = threadIdx.x & 31;
    const int wave  = threadIdx.x >> 5;
    const int tile  = blockIdx.x * 8 + wave;   // 8 waves per 256-thread block
    const int tileN = tile & 3;                // D/16 = 4 column tiles
    const int tileM = tile >> 2;

    const int mn    = lane & 15;   // row index of A / col index of B for this lane
    const int khalf = lane >> 4;   // 0: K pair {0,1}, 1: K pair {2,3}

    const float* arow = A + (tileM * 16 + mn) * D;     // A[tileM*16+mn][*]
    const float* bcol = B + tileN * 16 + mn;           // B[*][tileN*16+mn]

    v8f c = {};
    #pragma unroll
    for (int k = 0; k < D; k += 4) {
        v2f a, b;
        a.x = arow[k + 2 * khalf + 0];
        a.y = arow[k + 2 * khalf + 1];
        b.x = bcol[(k + 2 * khalf + 0) * D];
        b.y = bcol[(k + 2 * khalf + 1) * D];
        // 8 args: (neg_a, A, neg_b, B, c_mod, C, reuse_a, reuse_b)
        c = __builtin_amdgcn_wmma_f32_16x16x4_f32(false, a, false, b,
                                                  (short)0, c, false, false);
    }

    // Write back D tile: VGPR r -> row (tileM*16 + khalf*8 + r), col tileN*16+mn
    float* hout = H + (tileM * 16 + khalf * 8) * D + tileN * 16 + mn;
    #pragma unroll
    for (int r = 0; r < 8; ++r)
        hout[r * D] = c[r];
}

// ---------------------------------------------------------------------------
// Kernel 3: scatter-add.  One wave32 per edge: per-edge scalars are
// wave-uniform; each lane handles 2 consecutive features (float2 = b64 load,
// coalesced across the row), then 2 global f32 atomics into out[dst].
// h (16.8 MB) and out (16.8 MB) are both L2-resident (192 MB L2), so the
// gather and the atomic RMW are served at L2 bandwidth.
// ---------------------------------------------------------------------------
__global__ void gcn_scatter(const float* __restrict__ H,
                            const float* __restrict__ ew,
                            const int*   __restrict__ src,
                            const int*   __restrict__ dst,
                            float*       __restrict__ out) {
    const int gid  = blockIdx.x * blockDim.x + threadIdx.x;
    const int e    = gid >> 5;          // edge id (wave-uniform)
    const int lane = gid & 31;

    const int   s = src[e];
    const int   d = dst[e];
    const float w = ew[e];

    const v2f* hrow = (const v2f*)(H + (size_t)s * D);
    v2f hv = hrow[lane];                // features 2*lane, 2*lane+1

    float* orow = out + (size_t)d * D + 2 * lane;
    atomicAdd(orow + 0, hv.x * w);
    atomicAdd(orow + 1, hv.y * w);
}

// ---------------------------------------------------------------------------
extern "C" void kernel_launch(void* const* d_in, const int* in_sizes, int n_in,
                              void* d_out, int out_size, void* d_ws, size_t ws_size,
                              hipStream_t stream) {
    const float* support = (const float*)d_in[0];   // [N_NODES, D]
    const float* weight  = (const float*)d_in[1];   // [D, D]
    const float* bias    = (const float*)d_in[2];   // [D]
    const float* ew      = (const float*)d_in[3];   // [N_EDGES]
    const int*   src     = (const int*)d_in[4];     // [N_EDGES]
    const int*   dst     = (const int*)d_in[5];     // [N_EDGES]
    float*       out     = (float*)d_out;           // [N_NODES, D]
    float*       h       = (float*)d_ws;            // scratch: N_NODES*D*4 = 16.8 MB

    // 1) out = bias (broadcast)
    gcn_init_bias<<<(N_NODES * D) / 256, 256, 0, stream>>>(bias, out);

    // 2) h = support @ weight  (4096 * 4 = 16384 tiles, 8 waves/block)
    gcn_gemm_wmma<<<(N_NODES / 16) * (D / 16) / 8, 256, 0, stream>>>(support, weight, h);

    // 3) out[dst] += ew * h[src]  (one wave per edge)
    gcn_scatter<<<(unsigned)((size_t)N_EDGES * 32 / 256), 256, 0, stream>>>(h, ew, src, dst, out);
}